// Generator3DLUT_zero_231928234069
// MI455X (gfx1250) — compile-verified
//
#include <hip/hip_runtime.h>

typedef float v4f __attribute__((ext_vector_type(4)));
typedef float v2f __attribute__((ext_vector_type(2)));

#define LUT_DIM   33
#define LUT_DIM2  (LUT_DIM * LUT_DIM)
#define LUT_DIM3  (LUT_DIM * LUT_DIM * LUT_DIM)
#define IMG_HW    (1024 * 1024)
#define N_IMG     8
#define N_PIX     (N_IMG * IMG_HW)

// ---------------------------------------------------------------------------
// Pre-pass: repack planar LUT [3][33][33][33] -> interleaved [33^3][r,g,b,pad]
// so one global_load_b128 fetches all 3 channels of a corner.
// ---------------------------------------------------------------------------
__global__ __launch_bounds__(256) void repack_lut_kernel(
    const float* __restrict__ lut, v4f* __restrict__ lutp)
{
    int i = blockIdx.x * blockDim.x + threadIdx.x;
    if (i >= LUT_DIM3) return;
    v4f e;
    e.x = lut[i];
    e.y = lut[LUT_DIM3 + i];
    e.z = lut[2 * LUT_DIM3 + i];
    e.w = 0.0f;
    lutp[i] = e;
}

// ---------------------------------------------------------------------------
// Main kernel: 4 pixels per lane. Streamed image I/O uses non-temporal
// 128-bit loads/stores (keeps the ~575 KB repacked LUT resident in the 192 MB L2);
// LUT gathers are regular-temporal b128 loads.
// ---------------------------------------------------------------------------
__global__ __launch_bounds__(256) void lut3d_apply_kernel(
    const float* __restrict__ x, const v4f* __restrict__ lutp,
    float* __restrict__ out)
{
    const float inv_bin = (float)((double)(LUT_DIM - 1) / 1.000001);

    int g = blockIdx.x * blockDim.x + threadIdx.x;
    if (g >= N_PIX / 4) return;
    int i0 = g << 2;
    int n  = i0 >> 20;            // image index (HW = 2^20)
    int p  = i0 & (IMG_HW - 1);   // pixel offset inside a channel plane

    const float* bp = x + (size_t)n * 3 * IMG_HW + p;
    v4f r4 = __builtin_nontemporal_load((const v4f*)(bp));
    v4f g4 = __builtin_nontemporal_load((const v4f*)(bp + IMG_HW));
    v4f b4 = __builtin_nontemporal_load((const v4f*)(bp + 2 * IMG_HW));

    v4f or4, og4, ob4;
#pragma unroll
    for (int k = 0; k < 4; ++k) {
        float fr = r4[k] * inv_bin;
        float fg = g4[k] * inv_bin;
        float fb = b4[k] * inv_bin;
        int ir = (int)floorf(fr); ir = min(max(ir, 0), LUT_DIM - 2);
        int ig = (int)floorf(fg); ig = min(max(ig, 0), LUT_DIM - 2);
        int ib = (int)floorf(fb); ib = min(max(ib, 0), LUT_DIM - 2);
        float dr = fr - (float)ir;
        float dg = fg - (float)ig;
        float db = fb - (float)ib;

        int base = ib * LUT_DIM2 + ig * LUT_DIM + ir;
        v4f c000 = lutp[base];
        v4f c001 = lutp[base + 1];
        v4f c010 = lutp[base + LUT_DIM];
        v4f c011 = lutp[base + LUT_DIM + 1];
        v4f c100 = lutp[base + LUT_DIM2];
        v4f c101 = lutp[base + LUT_DIM2 + 1];
        v4f c110 = lutp[base + LUT_DIM2 + LUT_DIM];
        v4f c111 = lutp[base + LUT_DIM2 + LUT_DIM + 1];

        // trilinear lerp (vectorized over the 3 channels)
        v4f a0 = c000 + dr * (c001 - c000);
        v4f a1 = c010 + dr * (c011 - c010);
        v4f a2 = c100 + dr * (c101 - c100);
        v4f a3 = c110 + dr * (c111 - c110);
        v4f b0 = a0 + dg * (a1 - a0);
        v4f b1 = a2 + dg * (a3 - a2);
        v4f c  = b0 + db * (b1 - b0);

        or4[k] = c.x; og4[k] = c.y; ob4[k] = c.z;
    }

    float* op = out + (size_t)n * 3 * IMG_HW + p;
    __builtin_nontemporal_store(or4, (v4f*)(op));
    __builtin_nontemporal_store(og4, (v4f*)(op + IMG_HW));
    __builtin_nontemporal_store(ob4, (v4f*)(op + 2 * IMG_HW));
}

// ---------------------------------------------------------------------------
// Fallback (only if d_ws can't hold the repacked LUT): gather from the planar
// LUT directly, using 64-bit loads for the (ir, ir+1) contiguous pairs.
// ---------------------------------------------------------------------------
__global__ __launch_bounds__(256) void lut3d_apply_fallback_kernel(
    const float* __restrict__ x, const float* __restrict__ lut,
    float* __restrict__ out)
{
    const float inv_bin = (float)((double)(LUT_DIM - 1) / 1.000001);

    int g = blockIdx.x * blockDim.x + threadIdx.x;
    if (g >= N_PIX / 4) return;
    int i0 = g << 2;
    int n  = i0 >> 20;
    int p  = i0 & (IMG_HW - 1);

    const float* bp = x + (size_t)n * 3 * IMG_HW + p;
    v4f r4 = __builtin_nontemporal_load((const v4f*)(bp));
    v4f g4 = __builtin_nontemporal_load((const v4f*)(bp + IMG_HW));
    v4f b4 = __builtin_nontemporal_load((const v4f*)(bp + 2 * IMG_HW));

    v4f or4, og4, ob4;
#pragma unroll
    for (int k = 0; k < 4; ++k) {
        float fr = r4[k] * inv_bin;
        float fg = g4[k] * inv_bin;
        float fb = b4[k] * inv_bin;
        int ir = (int)floorf(fr); ir = min(max(ir, 0), LUT_DIM - 2);
        int ig = (int)floorf(fg); ig = min(max(ig, 0), LUT_DIM - 2);
        int ib = (int)floorf(fb); ib = min(max(ib, 0), LUT_DIM - 2);
        float dr = fr - (float)ir;
        float dg = fg - (float)ig;
        float db = fb - (float)ib;

        int base = ib * LUT_DIM2 + ig * LUT_DIM + ir;
        float oc[3];
#pragma unroll
        for (int c = 0; c < 3; ++c) {
            const float* lc = lut + (size_t)c * LUT_DIM3 + base;
            v2f p00 = *(const v2f*)(lc);
            v2f p01 = *(const v2f*)(lc + LUT_DIM);
            v2f p10 = *(const v2f*)(lc + LUT_DIM2);
            v2f p11 = *(const v2f*)(lc + LUT_DIM2 + LUT_DIM);
            float a0 = p00.x + dr * (p00.y - p00.x);
            float a1 = p01.x + dr * (p01.y - p01.x);
            float a2 = p10.x + dr * (p10.y - p10.x);
            float a3 = p11.x + dr * (p11.y - p11.x);
            float b0 = a0 + dg * (a1 - a0);
            float b1 = a2 + dg * (a3 - a2);
            oc[c] = b0 + db * (b1 - b0);
        }
        or4[k] = oc[0]; og4[k] = oc[1]; ob4[k] = oc[2];
    }

    float* op = out + (size_t)n * 3 * IMG_HW + p;
    __builtin_nontemporal_store(or4, (v4f*)(op));
    __builtin_nontemporal_store(og4, (v4f*)(op + IMG_HW));
    __builtin_nontemporal_store(ob4, (v4f*)(op + 2 * IMG_HW));
}

extern "C" void kernel_launch(void* const* d_in, const int* in_sizes, int n_in,
                              void* d_out, int out_size, void* d_ws, size_t ws_size,
                              hipStream_t stream) {
    const float* lut = (const float*)d_in[0];   // (3, 33, 33, 33) fp32
    const float* x   = (const float*)d_in[1];   // (8, 3, 1024, 1024) fp32
    float* out = (float*)d_out;                 // (8, 3, 1024, 1024) fp32

    const size_t lutp_bytes = (size_t)LUT_DIM3 * sizeof(v4f); // ~575 KB
    const int groups = N_PIX / 4;               // 2,097,152
    const int blocks = groups / 256;            // 8192 exactly

    if (ws_size >= lutp_bytes && d_ws != nullptr) {
        v4f* lutp = (v4f*)d_ws;
        repack_lut_kernel<<<(LUT_DIM3 + 255) / 256, 256, 0, stream>>>(lut, lutp);
        lut3d_apply_kernel<<<blocks, 256, 0, stream>>>(x, lutp, out);
    } else {
        lut3d_apply_fallback_kernel<<<blocks, 256, 0, stream>>>(x, lut, out);
    }
}